// CenterGroup_52879637348671
// MI455X (gfx1250) — compile-verified
//
#include <hip/hip_runtime.h>

typedef __attribute__((ext_vector_type(2))) float v2f;
typedef __attribute__((ext_vector_type(8))) float v8f;

#define NUM_GROUP   1024
#define GROUP_SIZE  32
#define NPTS        16384
#define BATCH       16
#define NCH         6
#define CHUNK       256
#define DIST_STRIDE 260   // 256 + 4 pad: row r+8 lands on banks +32 -> conflict-free WMMA stores

// One workgroup (128 threads = 4 wave32) handles one tile of 16 centers for one batch.
// Per 256-point chunk:
//   1) cooperative load+transform points -> LDS as (-2x,-2y,-2z,|p|^2)
//   2) each wave runs 4x V_WMMA_F32_16X16X4_F32: A=(cx,cy,cz,1) x B=(-2p,|p|^2)
//      giving dist' = |p|^2 - 2 c.p  (the per-row |c|^2 term never changes top-k order)
//   3) 8-thread teams per center prefilter dist' < current-32nd-best via ballot,
//      compact candidate indices deterministically (prefix popcount, ascending j),
//      leader thread does stable insertion into the sorted top-32 list in LDS.
// Finally gather the 6-channel points for the 32 winners and subtract the center.
__global__ __launch_bounds__(128)
void CenterGroup_kernel(const float* __restrict__ xyz,
                        const float* __restrict__ center,
                        float* __restrict__ out)
{
    __shared__ float ptsBuf[CHUNK * 4];            //  4 KB
    __shared__ float distBuf[16 * DIST_STRIDE];    // ~16.6 KB
    __shared__ float topD[16 * GROUP_SIZE];        //  2 KB (sorted ascending)
    __shared__ int   topI[16 * GROUP_SIZE];        //  2 KB
    __shared__ int   candIdx[16 * CHUNK];          // 16 KB
    __shared__ int   candCnt[16];

    const int tile = blockIdx.x;        // 0..63 : which 16-center tile
    const int b    = blockIdx.y;        // 0..15 : batch
    const int g0   = tile * 16;
    const int tid  = threadIdx.x;
    const int lane = tid & 31;
    const int wave = tid >> 5;
    const int c    = tid >> 3;          // team's center 0..15
    const int lt   = tid & 7;           // lane within 8-thread team

    // init top lists
    for (int t = tid; t < 16 * GROUP_SIZE; t += 128) {
        topD[t] = INFINITY;
        topI[t] = 0;
    }

    // Build A fragment: 16x4 f32. ISA layout: lanes 0-15 hold K=0 (v0), K=1 (v1);
    // lanes 16-31 hold K=2 (v0), K=3 (v1). Row M = lane & 15.
    const int    arow = lane & 15;
    const float* cp   = center + ((size_t)b * NUM_GROUP + g0 + arow) * 3;
    v2f afrag;
    if (lane < 16) { afrag.x = cp[0]; afrag.y = cp[1]; }
    else           { afrag.x = cp[2]; afrag.y = 1.0f; }

    const float* xb = xyz + (size_t)b * NPTS * NCH;

    __syncthreads();

    for (int chunk = 0; chunk < NPTS / CHUNK; ++chunk) {
        const int nbase = chunk * CHUNK;

        // ---- load + transform 256 points (2 per thread) ----
        #pragma unroll
        for (int j = 0; j < 2; ++j) {
            const int    pl = tid + j * 128;
            const float* pp = xb + (size_t)(nbase + pl) * NCH;   // 8-byte aligned
            float2 xy = *(const float2*)pp;
            float  z  = pp[2];
            float4 v;
            v.x = -2.0f * xy.x;
            v.y = -2.0f * xy.y;
            v.z = -2.0f * z;
            v.w = xy.x * xy.x + xy.y * xy.y + z * z;
            *(float4*)&ptsBuf[pl * 4] = v;                       // 16-byte aligned
        }
        __syncthreads();

        // ---- WMMA distance tiles: wave w does point sub-tiles 4w..4w+3 ----
        #pragma unroll
        for (int t = 0; t < 4; ++t) {
            const int p0 = (wave * 4 + t) * 16;
            const int pr = (p0 + (lane & 15)) * 4;
            v2f bfrag;
            if (lane < 16) { bfrag.x = ptsBuf[pr + 0]; bfrag.y = ptsBuf[pr + 1]; }
            else           { bfrag.x = ptsBuf[pr + 2]; bfrag.y = ptsBuf[pr + 3]; }

            v8f acc = {};
            acc = __builtin_amdgcn_wmma_f32_16x16x4_f32(
                false, afrag, false, bfrag, (short)0, acc, false, false);

            const int colBase = p0 + (lane & 15);
            const int rowBase = (lane >> 4) * 8;
            #pragma unroll
            for (int r = 0; r < 8; ++r)
                distBuf[(rowBase + r) * DIST_STRIDE + colBase] = acc[r];
        }
        __syncthreads();

        // ---- prefilter: 8 threads/center, ballot compaction (ascending j) ----
        {
            const float thr    = topD[c * GROUP_SIZE + (GROUP_SIZE - 1)];
            const int   tshift = (lane >> 3) << 3;   // team's bit offset in wave ballot
            int cnt = 0;
            #pragma unroll 4
            for (int i = 0; i < CHUNK / 8; ++i) {
                const int   j    = i * 8 + lt;
                const float dv   = distBuf[c * DIST_STRIDE + j];
                const bool  pred = dv < thr;
                const unsigned bal  = (unsigned)__ballot(pred);
                const unsigned team = (bal >> tshift) & 0xffu;
                if (pred) {
                    const int pos = __popc(team & ((1u << lt) - 1u));
                    candIdx[c * CHUNK + cnt + pos] = j;
                }
                cnt += __popc(team);
            }
            if (lt == 0) candCnt[c] = cnt;
        }
        __syncthreads();

        // ---- leader insertion: stable sorted insert into top-32 ----
        if (tid < 16) {
            float* td = &topD[tid * GROUP_SIZE];
            int*   ti = &topI[tid * GROUP_SIZE];
            const int cnt = candCnt[tid];
            for (int p = 0; p < cnt; ++p) {
                const int   j  = candIdx[tid * CHUNK + p];
                const float dv = distBuf[tid * DIST_STRIDE + j];
                if (dv < td[GROUP_SIZE - 1]) {
                    int k = GROUP_SIZE - 1;
                    while (k > 0 && td[k - 1] > dv) {   // strict: keeps ties stable
                        td[k] = td[k - 1];
                        ti[k] = ti[k - 1];
                        --k;
                    }
                    td[k] = dv;
                    ti[k] = nbase + j;
                }
            }
        }
        __syncthreads();
    }

    // ---- gather winners, subtract center from xyz channels ----
    for (int pair = tid; pair < 16 * GROUP_SIZE; pair += 128) {
        const int cc = pair >> 5;
        const int m  = pair & (GROUP_SIZE - 1);
        const int g  = g0 + cc;
        const int n  = topI[cc * GROUP_SIZE + m];
        const float* pp = xb + (size_t)n * NCH;
        const float* cg = center + ((size_t)b * NUM_GROUP + g) * 3;
        float* op = out + (((size_t)b * NUM_GROUP + g) * GROUP_SIZE + m) * NCH;
        float2 p01 = *(const float2*)(pp);
        float2 p23 = *(const float2*)(pp + 2);
        float2 p45 = *(const float2*)(pp + 4);
        float2 o01; o01.x = p01.x - cg[0]; o01.y = p01.y - cg[1];
        float2 o23; o23.x = p23.x - cg[2]; o23.y = p23.y;
        *(float2*)(op)     = o01;
        *(float2*)(op + 2) = o23;
        *(float2*)(op + 4) = p45;
    }
}

extern "C" void kernel_launch(void* const* d_in, const int* in_sizes, int n_in,
                              void* d_out, int out_size, void* d_ws, size_t ws_size,
                              hipStream_t stream) {
    (void)in_sizes; (void)n_in; (void)out_size; (void)d_ws; (void)ws_size;
    const float* xyz    = (const float*)d_in[0];   // (16, 16384, 6) f32
    const float* center = (const float*)d_in[1];   // (16, 1024, 3)  f32
    float*       out    = (float*)d_out;           // (16, 1024, 32, 6) f32

    dim3 grid(NUM_GROUP / 16, BATCH);
    dim3 block(128);
    CenterGroup_kernel<<<grid, block, 0, stream>>>(xyz, center, out);
}